// N3AggregationBase_34265249088171
// MI455X (gfx1250) — compile-verified
//
#include <hip/hip_runtime.h>
#include <hip/hip_bf16.h>

// Problem constants (fixed by the reference's setup_inputs()).
#define BDIM 4
#define NDIM 8192
#define MDIM 2048
#define ODIM 64      // candidates per query (and the GEMM inner dim)
#define EDIM 64      // embedding dim for distances
#define FDIM 128     // feature dim of x
#define KKNN 7       // relaxed top-k samples

// LDS row stride for the gathered x tile: 128 + 16 pad so the transposed
// A-fragment reads (lane-halves differ by 2 rows) land on disjoint bank groups
// (stride 144 words == 16 mod 64 -> halves are 32 banks apart).
#define LDSX_STRIDE 144

typedef float v2f __attribute__((ext_vector_type(2)));
typedef float v8f __attribute__((ext_vector_type(8)));
typedef int   v4i __attribute__((vector_size(16)));   // matches builtin prototype

// ---- optional CDNA5 async global->LDS gather path (ASYNCcnt) ----
#if defined(__has_builtin)
#  if __has_builtin(__builtin_amdgcn_global_load_async_to_lds_b128)
#    define USE_ASYNC_GATHER 1
#  endif
#endif
#ifndef USE_ASYNC_GATHER
#  define USE_ASYNC_GATHER 0
#endif

#if USE_ASYNC_GATHER
typedef __attribute__((address_space(1))) v4i* gv4p;  // global int4*
typedef __attribute__((address_space(3))) v4i* lv4p;  // LDS int4*
#endif

__device__ __forceinline__ float log1mexp_f(float v) {
  // log(1 - exp(v)), v <= 0 ; matches reference guard constants
  const float LOG_HALF = -0.69314718055994530942f;
  if (v < LOG_HALF) return log1pf(-expf(v));
  return logf(-expm1f(v) + 1e-7f);
}

__device__ __forceinline__ float block_reduce_max64(float* red, int t, float v) {
  __syncthreads();
  if (t < 64) red[t] = v;
  __syncthreads();
  if (t < 32) red[t] = fmaxf(red[t], red[t + 32]);
  __syncthreads();
  if (t < 16) red[t] = fmaxf(red[t], red[t + 16]);
  __syncthreads();
  if (t < 8)  red[t] = fmaxf(red[t], red[t + 8]);
  __syncthreads();
  if (t < 4)  red[t] = fmaxf(red[t], red[t + 4]);
  __syncthreads();
  if (t < 2)  red[t] = fmaxf(red[t], red[t + 2]);
  __syncthreads();
  if (t == 0) red[0] = fmaxf(red[0], red[1]);
  __syncthreads();
  float r = red[0];
  __syncthreads();
  return r;
}

__device__ __forceinline__ float block_reduce_sum64(float* red, int t, float v) {
  __syncthreads();
  if (t < 64) red[t] = v;
  __syncthreads();
  if (t < 32) red[t] += red[t + 32];
  __syncthreads();
  if (t < 16) red[t] += red[t + 16];
  __syncthreads();
  if (t < 8)  red[t] += red[t + 8];
  __syncthreads();
  if (t < 4)  red[t] += red[t + 4];
  __syncthreads();
  if (t < 2)  red[t] += red[t + 2];
  __syncthreads();
  if (t == 0) red[0] += red[1];
  __syncthreads();
  float r = red[0];
  __syncthreads();
  return r;
}

// One workgroup (256 threads = 8 wave32) per (b, m) pair.
__global__ void __launch_bounds__(256)
n3_agg_kernel(const float* __restrict__ x,   // [b,n,f]
              const float* __restrict__ xe,  // [b,n,e]
              const float* __restrict__ ye,  // [b,m,e]
              const float* __restrict__ lt,  // [1] log temperature
              const int*   __restrict__ I,   // [b,m,o] gather indices
              float* __restrict__ out)       // [b,m,f,K]
{
  __shared__ float lds_x[ODIM * LDSX_STRIDE]; // gathered x rows (transposed reads)
  __shared__ float lds_w[ODIM * 16];          // W, k padded 7 -> 16 with zeros
  __shared__ int   lds_idx[ODIM];
  __shared__ float red[256];

  const int t  = threadIdx.x;
  const int bi = blockIdx.x / MDIM;
  const int mi = blockIdx.x % MDIM;

  // ---- indices + zero-fill W padding ----
  if (t < ODIM) lds_idx[t] = I[((size_t)bi * MDIM + mi) * ODIM + t];
  for (int i = t; i < ODIM * 16; i += 256) lds_w[i] = 0.0f;
  __syncthreads();

  const int o = t >> 2;  // row 0..63 (4 threads per row)
  const int q = t & 3;   // quarter within the row

  // ---- kick off x-row gather into LDS (overlaps distance + softmax) ----
  {
    const float* xrow = x + ((size_t)bi * NDIM + lds_idx[o]) * FDIM;
    const int f0 = q * 32;                       // 32 floats per thread
#if USE_ASYNC_GATHER
#pragma unroll
    for (int c = 0; c < 8; ++c) {
      __builtin_amdgcn_global_load_async_to_lds_b128(
          (gv4p)(xrow + f0 + 4 * c),
          (lv4p)(&lds_x[o * LDSX_STRIDE + f0 + 4 * c]),
          0, 0);
    }
#else
#pragma unroll
    for (int c = 0; c < 8; ++c) {
      float4 v = *(const float4*)(xrow + f0 + 4 * c);
      *(float4*)(&lds_x[o * LDSX_STRIDE + f0 + 4 * c]) = v;
    }
#endif
  }

  // ---- distances: D[o] = -||xe[I[o]] - ye||^2 (partial per thread) ----
  {
    const float* xer = xe + ((size_t)bi * NDIM + lds_idx[o]) * EDIM + q * 16;
    const float* yer = ye + ((size_t)bi * MDIM + mi) * EDIM + q * 16;
    float s = 0.0f;
#pragma unroll
    for (int c = 0; c < 4; ++c) {
      float4 a = *(const float4*)(xer + 4 * c);
      float4 b = *(const float4*)(yer + 4 * c);
      float dx = a.x - b.x, dy = a.y - b.y, dz = a.z - b.z, dw = a.w - b.w;
      s += dx * dx + dy * dy + dz * dz + dw * dw;
    }
    red[t] = s;
  }
  __syncthreads();

  const float inv_temp = expf(-lt[0]);
  float logit = 0.0f;
  if (t < ODIM)
    logit = -(red[4 * t] + red[4 * t + 1] + red[4 * t + 2] + red[4 * t + 3]) * inv_temp;
  __syncthreads();

  // ---- K-step relaxed top-k (log-softmax + log1mexp update) ----
  for (int it = 0; it < KKNN; ++it) {
    float mx = block_reduce_max64(red, t, (t < ODIM) ? logit : -3.4e38f);
    float se = block_reduce_sum64(red, t, (t < ODIM) ? expf(logit - mx) : 0.0f);
    float w  = logit - (mx + logf(se));
    if (t < ODIM) {
      lds_w[t * 16 + it] = expf(w);
      logit += log1mexp_f(w);
    }
  }

  // ---- wait for the async gather, then converge for WMMA ----
#if USE_ASYNC_GATHER
#  if __has_builtin(__builtin_amdgcn_s_wait_asynccnt)
  __builtin_amdgcn_s_wait_asynccnt(0);
#  else
  asm volatile("s_wait_asynccnt 0" ::: "memory");
#  endif
#endif
  __syncthreads();

  // ---- aggregation GEMM: z[f,k] = sum_o xg[o,f] * W[o,k] via WMMA f32 16x16x4.
  // Wave wv owns f-tile [16*wv, 16*wv+16); accumulates 16 K-steps of 4.
  // A frag: lane L, vgpr v -> (M = L%16, K = 2*(L/16)+v)
  // B frag: lane L, vgpr v -> (K = 2*(L/16)+v, N = L%16)
  const int wv   = t >> 5;
  const int lane = t & 31;
  const int half = lane >> 4;
  const int fr   = lane & 15;
  const int f0   = wv * 16;

  v8f acc = {};
#pragma unroll
  for (int s = 0; s < 16; ++s) {
    const int o0 = 4 * s + 2 * half;
    v2f a, b;
    a.x = lds_x[o0 * LDSX_STRIDE + f0 + fr];
    a.y = lds_x[(o0 + 1) * LDSX_STRIDE + f0 + fr];
    b.x = lds_w[o0 * 16 + fr];
    b.y = lds_w[(o0 + 1) * 16 + fr];
    acc = __builtin_amdgcn_wmma_f32_16x16x4_f32(
        /*neg_a=*/false, a, /*neg_b=*/false, b,
        /*c_mod=*/(short)0, acc, /*reuse_a=*/false, /*reuse_b=*/false);
  }

  // ---- store z[b,m,f,k]: C lane L, vgpr v -> (f = f0 + 8*half + v, k = L%16) ----
  if (fr < KKNN) {
    union { v8f v; float f[8]; } u;
    u.v = acc;
    float* zb = out + ((size_t)bi * MDIM + mi) * (size_t)(FDIM * KKNN);
#pragma unroll
    for (int v = 0; v < 8; ++v) {
      const int f = f0 + 8 * half + v;
      zb[(size_t)f * KKNN + fr] = u.f[v];
    }
  }
}

extern "C" void kernel_launch(void* const* d_in, const int* in_sizes, int n_in,
                              void* d_out, int out_size, void* d_ws, size_t ws_size,
                              hipStream_t stream) {
  const float* x  = (const float*)d_in[0];
  const float* xe = (const float*)d_in[1];
  const float* ye = (const float*)d_in[2];
  const float* lt = (const float*)d_in[3];
  const int*   I  = (const int*)d_in[4];   // harness passes integer inputs as int32
  float* out = (float*)d_out;

  dim3 grid(BDIM * MDIM);
  dim3 block(256);
  hipLaunchKernelGGL(n3_agg_kernel, grid, block, 0, stream, x, xe, ye, lt, I, out);
}